// TaskAlignedAssigner_21646635172366
// MI455X (gfx1250) — compile-verified
//
#include <hip/hip_runtime.h>
#include <hip/hip_bf16.h>
#include <math.h>

#define Bb 16
#define Nn 64
#define Aa 8400
#define Cc 20
#define TOPKK 13
#define NIT 66   // ceil(8400/128)

typedef float v2f __attribute__((ext_vector_type(2)));
typedef float v8f __attribute__((ext_vector_type(8)));
typedef unsigned long long u64;

// CIoU clipped to >=0; dd = atan(w2/(h2+eps)) - atan(w1/(h1+eps)) precomputed by caller.
__device__ __forceinline__ float ciou_clip(float gx1,float gy1,float gx2,float gy2,
                                           float px1,float py1,float px2,float py2,
                                           float dd) {
  const float eps = 1e-7f;
  float w1=gx2-gx1, h1=gy2-gy1;
  float w2=px2-px1, h2=py2-py1;
  float iw = fmaxf(fminf(gx2,px2)-fmaxf(gx1,px1), 0.f);
  float ih = fmaxf(fminf(gy2,py2)-fmaxf(gy1,py1), 0.f);
  float inter = iw*ih;
  float uni = w1*h1 + w2*h2 - inter + eps;
  float iou = inter/uni;
  float cw = fmaxf(gx2,px2)-fminf(gx1,px1);
  float ch = fmaxf(gy2,py2)-fminf(gy1,py1);
  float c2 = cw*cw + ch*ch + eps;
  float dx = px1+px2-gx1-gx2, dy = py1+py2-gy1-gy2;
  float rho2 = (dx*dx+dy*dy)*0.25f;
  const float k4pi2 = 4.0f/(3.14159265358979323846f*3.14159265358979323846f);
  float v = k4pi2*dd*dd;
  float alpha = v/(v - iou + 1.f + eps);
  return fmaxf(iou - (rho2/c2 + v*alpha), 0.f);
}

__device__ __forceinline__ void async_copy16(unsigned lds_off, const void* gptr) {
  unsigned long long ga = (unsigned long long)(uintptr_t)gptr;
  asm volatile("global_load_async_to_lds_b128 %0, %1, off"
               :: "v"(lds_off), "v"(ga) : "memory");
}

// K1: fused score-WMMA + CIoU + align + per-(b,n) top-13 + mask_pos bit scatter.
// grid = Bb*4 blocks (16 gt rows each), 256 threads (8 waves, wave32).
__global__ __launch_bounds__(256) void k1_topk(
    const float* __restrict__ pd_scores, const float* __restrict__ pd_bboxes,
    const float* __restrict__ anc, const int* __restrict__ gt_labels,
    const float* __restrict__ gt_bboxes, const float* __restrict__ mask_gt,
    u64* __restrict__ maskpos, int* __restrict__ topk_idx, float* __restrict__ topk_val)
{
  __shared__ float  s_tile[16*132];          // align tile: 16 rows x 128 anchors (pitch 132)
  __shared__ u64    s_lists[256*TOPKK];      // per-thread sorted top-13 keys
  __shared__ float  s_pb[2][128*4];          // double-buffered pd boxes (async staged)
  __shared__ float4 s_gt[16];
  __shared__ float  s_atan[16];
  __shared__ float  s_valid[16];
  __shared__ int    s_lab[16];

  const int b    = blockIdx.x >> 2;
  const int n0   = (blockIdx.x & 3) * 16;
  const int t    = threadIdx.x;
  const int lane = t & 31;
  const int wv   = t >> 5;            // wave 0..7 -> 16-anchor subtile
  const int lh   = lane >> 4;         // half-wave select
  const int lc   = lane & 15;         // col within subtile / gt row for A-matrix
  const int row  = t >> 4;            // top-k ownership: row 0..15
  const int slot = t & 15;            // 16 scanners per row

  if (t < 16) {
    int n = n0 + t;
    float4 g = ((const float4*)gt_bboxes)[b*Nn + n];
    s_gt[t]    = g;
    s_atan[t]  = atanf((g.z-g.x)/((g.w-g.y)+1e-7f));
    s_valid[t] = mask_gt[b*Nn + n];
    s_lab[t]   = gt_labels[b*Nn + n];
  }
  for (int i = 0; i < TOPKK; ++i) s_lists[t*TOPKK + i] = 0ull;

  // prefetch anchor tile 0 into LDS via async-to-LDS DMA path
  if (t < 128) {
    int a0 = min(t, Aa-1);
    async_copy16((unsigned)(uintptr_t)&s_pb[0][t*4],
                 pd_bboxes + ((size_t)b*Aa + (size_t)a0)*4);
  }
  __syncthreads();

  for (int it = 0; it < NIT; ++it) {
    // pipeline: kick tile it+1, wait until only it+1 is outstanding -> tile it ready
    if (t < 128) {
      int nit = min(it+1, NIT-1);
      int a0  = min(nit*128 + t, Aa-1);
      async_copy16((unsigned)(uintptr_t)&s_pb[(it+1)&1][t*4],
                   pd_bboxes + ((size_t)b*Aa + (size_t)a0)*4);
      asm volatile("s_wait_asynccnt 0x1" ::: "memory");
    }
    __syncthreads();

    const int aIdx = it*128 + wv*16 + lc;
    const int aCl  = min(aIdx, Aa-1);

    // scores[16 gts x 16 anchors] = onehot(labels) x pd_scores^T : 5x V_WMMA_F32_16X16X4_F32
    v8f acc = {0.f,0.f,0.f,0.f,0.f,0.f,0.f,0.f};
    {
      const float* sp = pd_scores + ((size_t)b*Aa + (size_t)aCl)*Cc;
      const int lab = s_lab[lc];
      #pragma unroll
      for (int k0 = 0; k0 < Cc; k0 += 4) {
        const int kk = k0 + lh*2;     // lanes16-31 hold K=+2 pair (A and B 16x4/4x16 layout)
        v2f av, bv;
        av.x = (lab == kk    ) ? 1.f : 0.f;
        av.y = (lab == kk + 1) ? 1.f : 0.f;
        bv.x = sp[kk];
        bv.y = sp[kk+1];
        acc = __builtin_amdgcn_wmma_f32_16x16x4_f32(false, av, false, bv,
                                                    (short)0, acc, false, false);
      }
    }

    // one pd box + anc point per lane, reused for 8 gt rows (D layout rows j, j+8)
    const float* pbp = &s_pb[it & 1][(wv*16 + lc)*4];
    float px1=pbp[0], py1=pbp[1], px2=pbp[2], py2=pbp[3];
    float ax = anc[aCl*2], ay = anc[aCl*2+1];
    float at2 = atanf((px2-px1)/((py2-py1)+1e-7f));
    #pragma unroll
    for (int j = 0; j < 8; ++j) {
      int m = j + lh*8;
      float4 g = s_gt[m];
      float d  = fminf(fminf(ax-g.x, ay-g.y), fminf(g.z-ax, g.w-ay));
      float ov = ciou_clip(g.x,g.y,g.z,g.w, px1,py1,px2,py2, at2 - s_atan[m]);
      float p2 = ov*ov;
      float align = acc[j] * p2*p2*p2;           // score^1 * ciou^6
      bool valid = (d > 1e-9f) && (s_valid[m] > 0.f);
      s_tile[m*132 + wv*16 + lc] = valid ? align : 0.f;
    }
    __syncthreads();

    // per-thread top-13 insertion; key packs value + ~idx (tie-break lowest index)
    u64* lst = &s_lists[t*TOPKK];
    #pragma unroll
    for (int q = 0; q < 8; ++q) {
      int col = slot*8 + q;
      int aG  = it*128 + col;
      if (aG < Aa) {
        float val = s_tile[row*132 + col];
        u64 key = ((u64)__float_as_uint(val) << 32) | (unsigned)(~(unsigned)aG);
        if (key > lst[TOPKK-1]) {
          int p = TOPKK-1;
          while (p > 0 && lst[p-1] < key) { lst[p] = lst[p-1]; --p; }
          lst[p] = key;
        }
      }
    }
    __syncthreads();
  }

  // merge 16 lists per row: 13 rounds of half-wave (16-lane) shfl_xor max
  int cur = 0;
  u64 mywin = 0ull;
  const unsigned gsh = lane & 16;
  u64* lst = &s_lists[t*TOPKK];
  for (int r = 0; r < TOPKK; ++r) {
    u64 head = (cur < TOPKK) ? lst[cur] : 0ull;
    u64 mx = head;
    #pragma unroll
    for (int o = 1; o < 16; o <<= 1) {
      u64 oth = __shfl_xor(mx, o, 16);
      mx = (oth > mx) ? oth : mx;
    }
    u64 bal = __ballot(head == mx);
    unsigned grp = (unsigned)(bal >> gsh) & 0xFFFFu;
    int win = (grp ? (__ffs(grp) - 1) : 0) + (int)gsh;
    if (lane == win) ++cur;               // winner pops its head
    if ((lane & 15) == r) mywin = mx;     // lane r of group keeps r-th winner
  }

  // emit top-k entries + scatter mask_pos bits (topk AND in_gts AND mask_gt)
  if (slot < TOPKK) {
    unsigned aG = ~(unsigned)(mywin & 0xFFFFFFFFull);
    float val = __uint_as_float((unsigned)(mywin >> 32));
    int n = n0 + row;
    int oidx = (b*Nn + n)*TOPKK + slot;
    bool ok = (mywin != 0ull) && (aG < (unsigned)Aa);
    topk_idx[oidx] = ok ? (int)aG : -1;
    topk_val[oidx] = ok ? val : 0.f;
    if (ok && s_valid[row] > 0.f) {
      float axp = anc[aG*2], ayp = anc[aG*2+1];
      float4 g = s_gt[row];
      float d = fminf(fminf(axp-g.x, ayp-g.y), fminf(g.z-axp, g.w-ayp));
      if (d > 1e-9f)
        atomicOr(&maskpos[(size_t)b*Aa + aG], 1ull << n);
    }
  }
}

// K2: dedup multi-assigned anchors -> keep only argmax_n(overlaps) bit.
__global__ __launch_bounds__(256) void k2_dedup(
    const float* __restrict__ pd_bboxes, const float* __restrict__ anc,
    const float* __restrict__ gt_bboxes, const float* __restrict__ mask_gt,
    const u64* __restrict__ maskpos, u64* __restrict__ w2)
{
  int gid = blockIdx.x*blockDim.x + threadIdx.x;
  if (gid >= Bb*Aa) return;
  int b = gid / Aa, a = gid - b*Aa;
  u64 w = maskpos[gid];
  if (__popcll(w) > 1) {
    float4 pb = ((const float4*)pd_bboxes)[gid];
    float ax = anc[a*2], ay = anc[a*2+1];
    float at2 = atanf((pb.z-pb.x)/((pb.w-pb.y)+1e-7f));
    float best = -1.f; int bestn = 0;
    for (int n = 0; n < Nn; ++n) {
      float4 g = ((const float4*)gt_bboxes)[b*Nn + n];
      float ov = 0.f;
      if (mask_gt[b*Nn + n] > 0.f) {
        float d = fminf(fminf(ax-g.x, ay-g.y), fminf(g.z-ax, g.w-ay));
        if (d > 1e-9f) {
          float atg = atanf((g.z-g.x)/((g.w-g.y)+1e-7f));
          ov = ciou_clip(g.x,g.y,g.z,g.w, pb.x,pb.y,pb.z,pb.w, at2-atg);
        }
      }
      if (ov > best) { best = ov; bestn = n; }   // first-max like jnp.argmax
    }
    w &= (1ull << bestn);
  }
  w2[gid] = w;
}

// K3: per (b,n): rnorm = pos_ov / (pos_align + EPS) over surviving top-k entries.
__global__ __launch_bounds__(256) void k3_norm(
    const float* __restrict__ pd_bboxes, const float* __restrict__ gt_bboxes,
    const u64* __restrict__ w2, const int* __restrict__ topk_idx,
    const float* __restrict__ topk_val, float* __restrict__ rnorm)
{
  int gid = blockIdx.x*blockDim.x + threadIdx.x;
  if (gid >= Bb*Nn) return;
  int b = gid / Nn, n = gid - b*Nn;
  float4 g = ((const float4*)gt_bboxes)[gid];
  float atg = atanf((g.z-g.x)/((g.w-g.y)+1e-7f));
  float pa = 0.f, po = 0.f;
  for (int s = 0; s < TOPKK; ++s) {
    int a = topk_idx[gid*TOPKK + s];
    if (a < 0) continue;
    if ((w2[(size_t)b*Aa + a] >> n) & 1ull) {
      pa = fmaxf(pa, topk_val[gid*TOPKK + s]);
      float4 pb = ((const float4*)pd_bboxes)[(size_t)b*Aa + a];
      float at2 = atanf((pb.z-pb.x)/((pb.w-pb.y)+1e-7f));
      po = fmaxf(po, ciou_clip(g.x,g.y,g.z,g.w, pb.x,pb.y,pb.z,pb.w, at2-atg));
    }
  }
  rnorm[gid] = po / (pa + 1e-9f);
}

// K4: per (b,a) emit labels, boxes, one-hot*norm scores, fg, target_gt_idx.
__global__ __launch_bounds__(256) void k4_out(
    const int* __restrict__ gt_labels, const float* __restrict__ gt_bboxes,
    const u64* __restrict__ w2, const int* __restrict__ topk_idx,
    const float* __restrict__ topk_val, const float* __restrict__ rnorm,
    float* __restrict__ out)
{
  int gid = blockIdx.x*blockDim.x + threadIdx.x;
  if (gid >= Bb*Aa) return;
  int b = gid / Aa, a = gid - b*Aa;
  u64 w = w2[gid];
  bool fg = (w != 0ull);
  int n = fg ? (__ffsll((unsigned long long)w) - 1) : 0;  // argmax(first) of 0/1 mask
  float nm = 0.f;
  if (fg) {
    int base = (b*Nn + n)*TOPKK;
    for (int s = 0; s < TOPKK; ++s)
      if (topk_idx[base + s] == a) { nm = topk_val[base + s]; break; }
    nm *= rnorm[b*Nn + n];
  }
  int lab = gt_labels[b*Nn + n]; if (lab < 0) lab = 0;
  float4 g = ((const float4*)gt_bboxes)[b*Nn + n];
  const size_t BA = (size_t)Bb*Aa;
  out[gid] = (float)lab;
  float* ob = out + BA + (size_t)gid*4;
  ob[0]=g.x; ob[1]=g.y; ob[2]=g.z; ob[3]=g.w;
  float* os = out + 5*BA + (size_t)gid*Cc;
  #pragma unroll
  for (int c = 0; c < Cc; ++c) os[c] = (fg && c == lab) ? nm : 0.f;
  out[25*BA + gid] = fg ? 1.f : 0.f;
  out[26*BA + gid] = (float)n;
}

extern "C" void kernel_launch(void* const* d_in, const int* in_sizes, int n_in,
                              void* d_out, int out_size, void* d_ws, size_t ws_size,
                              hipStream_t stream) {
  (void)in_sizes; (void)n_in; (void)out_size;
  const float* pd_scores = (const float*)d_in[0];
  const float* pd_bboxes = (const float*)d_in[1];
  const float* anc       = (const float*)d_in[2];
  const int*   gt_labels = (const int*)d_in[3];
  const float* gt_bboxes = (const float*)d_in[4];
  const float* mask_gt   = (const float*)d_in[5];
  float* out = (float*)d_out;

  const size_t BA = (size_t)Bb*Aa;
  const size_t BN = (size_t)Bb*Nn;
  char* ws = (char*)d_ws;
  u64*   maskpos = (u64*)ws;
  u64*   w2      = (u64*)(ws + BA*8);
  int*   tidx    = (int*)(ws + 2*BA*8);
  float* tval    = (float*)(ws + 2*BA*8 + BN*TOPKK*4);
  float* rnorm   = (float*)(ws + 2*BA*8 + 2*BN*TOPKK*4);
  if (ws_size < 2*BA*8 + 2*BN*TOPKK*4 + BN*4) return;

  hipMemsetAsync(maskpos, 0, BA*8, stream);
  k1_topk<<<Bb*4, 256, 0, stream>>>(pd_scores, pd_bboxes, anc, gt_labels,
                                    gt_bboxes, mask_gt, maskpos, tidx, tval);
  k2_dedup<<<(int)((BA + 255)/256), 256, 0, stream>>>(pd_bboxes, anc, gt_bboxes,
                                                      mask_gt, maskpos, w2);
  k3_norm<<<(int)((BN + 255)/256), 256, 0, stream>>>(pd_bboxes, gt_bboxes, w2,
                                                     tidx, tval, rnorm);
  k4_out<<<(int)((BA + 255)/256), 256, 0, stream>>>(gt_labels, gt_bboxes, w2,
                                                    tidx, tval, rnorm, out);
}